// SGCNet_28991029248693
// MI455X (gfx1250) — compile-verified
//
#include <hip/hip_runtime.h>
#include <cmath>

// ---------------------------------------------------------------------------
// SGC on MI455X (gfx1250, wave32).
//   1) edge gate: ew = sigmoid(w) * (|w|>0); deg[col] += ew  (self-loop adds 1)
//   2) dinv = rsqrt(deg)
//   3) h0 = x * (sigmoid(xw) * (|xw|>0))                     [feature gate]
//   4) 2 hops: h' = dinv[i]^2 * h  (self loop)  +  scatter over edges
//              h'[col] += dinv[row]*ew*dinv[col] * h[row]    (f32 atomics)
//   5) out = h @ W^T + b  via V_WMMA_F32_16X16X4_F32, then per-row log_softmax
// GEMM is memory-bound (AI ~21 FLOP/B vs 23.3 TB/s HBM, all of h is L2-resident
// in the 192MB L2), so fp32 WMMA is chosen: exact precision at zero cost.
// ---------------------------------------------------------------------------

#define N_NODES 100000
#define N_EDGES 600000
#define N_FEAT  128
#define N_CLASS 64
#define K_HOPS  2

typedef __attribute__((ext_vector_type(2))) float v2f;
typedef __attribute__((ext_vector_type(4))) float v4f;
typedef __attribute__((ext_vector_type(8))) float v8f;

__device__ __forceinline__ void atomic_add_f32(float* p, float v) {
  // Lowers to global_atomic_add_f32 (agent scope, relaxed) -- no CAS loop.
  __hip_atomic_fetch_add(p, v, __ATOMIC_RELAXED, __HIP_MEMORY_SCOPE_AGENT);
}

__device__ __forceinline__ float sigmoid_gated(float w) {
  float m = (fabsf(w) > 0.0f) ? 1.0f : 0.0f;
  return m / (1.0f + __expf(-w));
}

// --- step 1a: self-loop contribution to degree ------------------------------
__global__ void k_init_deg(float* __restrict__ deg) {
  int i = blockIdx.x * blockDim.x + threadIdx.x;
  if (i < N_NODES) deg[i] = 1.0f;
}

// --- step 1b: gate edges, accumulate degree ---------------------------------
__global__ void k_edge_gate(const float* __restrict__ ewt,
                            const long long* __restrict__ ei,   // [2, E] int64
                            float* __restrict__ ew, float* __restrict__ deg) {
  int e = blockIdx.x * blockDim.x + threadIdx.x;
  if (e >= N_EDGES) return;
  float s = sigmoid_gated(ewt[e]);
  ew[e] = s;
  if (s != 0.0f) {
    int col = (int)ei[N_EDGES + e];
    atomic_add_f32(deg + col, s);
  }
}

// --- step 2: deg -> D^-1/2 in place -----------------------------------------
__global__ void k_dinv(float* __restrict__ deg) {
  int i = blockIdx.x * blockDim.x + threadIdx.x;
  if (i >= N_NODES) return;
  float d = deg[i];
  deg[i] = (d > 0.0f) ? rsqrtf(fmaxf(d, 1e-30f)) : 0.0f;
}

// --- step 3: feature gating, float4 vectorized ------------------------------
__global__ void k_feat_gate(const float* __restrict__ x,
                            const float* __restrict__ xw,
                            float* __restrict__ h) {
  int idx = blockIdx.x * blockDim.x + threadIdx.x;       // over N*F/4
  if (idx >= N_NODES * N_FEAT / 4) return;
  int f0 = (idx * 4) & (N_FEAT - 1);
  v4f xv = *(const v4f*)(x + (size_t)idx * 4);
  v4f g;
#pragma unroll
  for (int i = 0; i < 4; ++i) g[i] = sigmoid_gated(xw[f0 + i]);
  *(v4f*)(h + (size_t)idx * 4) = xv * g;
}

// --- step 4a: self-loop term initializes the output of a hop ----------------
__global__ void k_selfloop(const float* __restrict__ dinv,
                           const float* __restrict__ hin,
                           float* __restrict__ hout) {
  int idx = blockIdx.x * blockDim.x + threadIdx.x;       // over N*F/4
  if (idx >= N_NODES * N_FEAT / 4) return;
  int n = idx >> 5;                                      // / (F/4 = 32)
  float d = dinv[n];
  v4f v = *(const v4f*)(hin + (size_t)idx * 4);
  *(v4f*)(hout + (size_t)idx * 4) = v * (d * d);
}

// --- step 4b: edge scatter; one wave32 per edge, float4 per lane ------------
__global__ void k_scatter(const long long* __restrict__ ei,
                          const float* __restrict__ ew,
                          const float* __restrict__ dinv,
                          const float* __restrict__ hin,
                          float* __restrict__ hout) {
  int e = (int)((blockIdx.x * (unsigned)blockDim.x + threadIdx.x) >> 5);
  int lane = threadIdx.x & 31;
  if (e >= N_EDGES) return;
  float w = ew[e];
  if (w == 0.0f) return;
  int row = (int)ei[e];
  int col = (int)ei[N_EDGES + e];
  float norm = dinv[row] * w * dinv[col];
  if (norm == 0.0f) return;
  v4f src = *(const v4f*)(hin + (size_t)row * N_FEAT + lane * 4);
  float* dst = hout + (size_t)col * N_FEAT + lane * 4;
  atomic_add_f32(dst + 0, norm * src.x);
  atomic_add_f32(dst + 1, norm * src.y);
  atomic_add_f32(dst + 2, norm * src.z);
  atomic_add_f32(dst + 3, norm * src.w);
}

// --- step 5: WMMA GEMM (fp32 16x16x4) + bias + log_softmax ------------------
// One wave per 16-row tile. 4 accumulators cover N_CLASS=64 (4 x 16-col tiles).
// A layout (ISA 7.12.2, 32-bit A 16x4): lane l, vgpr j -> A[M=l%16, K=2*(l/16)+j]
// B layout assumed symmetric:          lane l, vgpr j -> B[K=2*(l/16)+j, N=l%16]
// B[k][n] = lin_w[n][k]  (out = h @ W^T).
#define GEMM_WAVES 4
__global__ void k_gemm_lsm(const float* __restrict__ h,
                           const float* __restrict__ W,     // [64,128]
                           const float* __restrict__ bias,  // [64]
                           float* __restrict__ out) {       // [N,64]
  __shared__ float smem[GEMM_WAVES * 16 * 65];              // stride 65: no bank conflicts
  const int wave = threadIdx.x >> 5;
  const int lane = threadIdx.x & 31;
  const int rowTile = blockIdx.x * GEMM_WAVES + wave;
  if (rowTile >= N_NODES / 16) return;                      // wave-uniform (EXEC stays full)

  const int m  = lane & 15;          // M for A, N for B
  const int kh = (lane >> 4) * 2;    // K sub-offset within a k-step
  const float* hrow = h + ((size_t)rowTile * 16 + m) * N_FEAT;

  v8f acc[4] = {v8f{}, v8f{}, v8f{}, v8f{}};
#pragma unroll 4
  for (int k = 0; k < N_FEAT; k += 4) {
    v2f a = *(const v2f*)(hrow + k + kh);
#pragma unroll
    for (int t = 0; t < 4; ++t) {
      v2f b = *(const v2f*)(W + (size_t)(t * 16 + m) * N_FEAT + k + kh);
      acc[t] = __builtin_amdgcn_wmma_f32_16x16x4_f32(
          /*neg_a=*/false, a, /*neg_b=*/false, b,
          /*c_mod=*/(short)0, acc[t], /*reuse_a=*/false, /*reuse_b=*/false);
    }
  }

  // Stage 16x64 tile to LDS. C/D layout: vgpr v, lane l -> row v+8*(l/16), col l%16.
  float* tile = smem + wave * 16 * 65;
  const int rhalf = (lane >> 4) * 8;
#pragma unroll
  for (int t = 0; t < 4; ++t)
#pragma unroll
    for (int v = 0; v < 8; ++v)
      tile[(v + rhalf) * 65 + t * 16 + m] = acc[t][v];

  // Own-wave LDS RAW: compiler tracks DScnt, but make it explicit (CDNA5 ISA
  // split wait counters, CDNA5 ISA).
  asm volatile("s_wait_dscnt 0x0" ::: "memory");

  // Per-row bias + log_softmax: lane r (<16) owns row r of the tile.
  if (lane < 16) {
    float* rowp = tile + lane * 65;
    float mx = -INFINITY;
#pragma unroll 8
    for (int c = 0; c < N_CLASS; ++c) {
      float v = rowp[c] + bias[c];
      rowp[c] = v;
      mx = fmaxf(mx, v);
    }
    float s = 0.0f;
#pragma unroll 8
    for (int c = 0; c < N_CLASS; ++c) s += __expf(rowp[c] - mx);
    float lse = mx + __logf(s);
    float* op = out + ((size_t)rowTile * 16 + lane) * N_CLASS;
#pragma unroll 8
    for (int c = 0; c < N_CLASS; ++c) op[c] = rowp[c] - lse;
  }
}

// ---------------------------------------------------------------------------
extern "C" void kernel_launch(void* const* d_in, const int* in_sizes, int n_in,
                              void* d_out, int out_size, void* d_ws, size_t ws_size,
                              hipStream_t stream) {
  const float*      x    = (const float*)d_in[0];        // [N,128]
  const long long*  ei   = (const long long*)d_in[1];    // [2,E] int64
  const float*      ewt  = (const float*)d_in[2];        // [E]
  const float*      xw   = (const float*)d_in[3];        // [128]
  const float*      linw = (const float*)d_in[4];        // [64,128]
  const float*      linb = (const float*)d_in[5];        // [64]
  float*            out  = (float*)d_out;                // [N,64]

  // Workspace layout (floats): ew[E] | dinv[N] | hA[N*F] | hB[N*F]  (~105 MB)
  float* ew   = (float*)d_ws;
  float* dinv = ew + N_EDGES;
  float* hA   = dinv + N_NODES;
  float* hB   = hA + (size_t)N_NODES * N_FEAT;

  const int NT = 256;
  const int NF4 = N_NODES * N_FEAT / 4;

  k_init_deg <<<(N_NODES + NT - 1) / NT, NT, 0, stream>>>(dinv);
  k_edge_gate<<<(N_EDGES + NT - 1) / NT, NT, 0, stream>>>(ewt, ei, ew, dinv);
  k_dinv     <<<(N_NODES + NT - 1) / NT, NT, 0, stream>>>(dinv);
  k_feat_gate<<<(NF4 + NT - 1) / NT, NT, 0, stream>>>(x, xw, hA);

  float* src = hA;
  float* dst = hB;
  for (int hop = 0; hop < K_HOPS; ++hop) {
    k_selfloop<<<(NF4 + NT - 1) / NT, NT, 0, stream>>>(dinv, src, dst);
    // one wave32 per edge, 8 waves / 256-thread block
    k_scatter <<<(N_EDGES + 7) / 8, NT, 0, stream>>>(ei, ew, dinv, src, dst);
    float* t = src; src = dst; dst = t;
  }

  const int nTiles = N_NODES / 16;                       // 6250
  k_gemm_lsm<<<(nTiles + GEMM_WAVES - 1) / GEMM_WAVES, GEMM_WAVES * 32, 0, stream>>>(
      src, linw, linb, out);
}